// DualSpatialBlock_15341623181911
// MI455X (gfx1250) — compile-verified
//
#include <hip/hip_runtime.h>
#include <hip/hip_bf16.h>

typedef float v2f __attribute__((ext_vector_type(2)));
typedef float v8f __attribute__((ext_vector_type(8)));

#define HEADS 2
#define HDIM 32
#define NEG_SLOPE 0.2f

__device__ __forceinline__ v8f wmma_f32_k4(v2f a, v2f b, v8f c) {
  // D = A(16x4,f32) * B(4x16,f32) + C(16x16,f32)
  return __builtin_amdgcn_wmma_f32_16x16x4_f32(false, a, false, b, (short)0, c,
                                               false, false);
}

__device__ __forceinline__ void atomicMaxF(float* addr, float v) {
  if (v >= 0.0f)
    atomicMax((int*)addr, __float_as_int(v));
  else
    atomicMin((unsigned int*)addr, __float_as_uint(v));
}

// ---------------------------------------------------------------------------
// init: mx = -inf, [den|gat|dense_agg] = 0
// ---------------------------------------------------------------------------
__global__ void init_ws_kernel(float* mx, int nmx, float* zbuf, int nz) {
  int i = blockIdx.x * blockDim.x + threadIdx.x;
  if (i < nmx) ((unsigned int*)mx)[i] = 0xFF800000u;  // -inf
  if (i < nz) zbuf[i] = 0.0f;
}

// ---------------------------------------------------------------------------
// precompute: Wfused = Wd @ Wf[64:128,:];  cvec = bias_gat@Wf[0:64] + bd@Wf[64:128] + bf
// ---------------------------------------------------------------------------
__global__ void precompute_kernel(const float* __restrict__ Wd,
                                  const float* __restrict__ Wf,
                                  const float* __restrict__ bias_gat,
                                  const float* __restrict__ bd,
                                  const float* __restrict__ bf,
                                  float* __restrict__ Wfused,
                                  float* __restrict__ cvec) {
  int tid = blockIdx.x * blockDim.x + threadIdx.x;
  if (tid < 64 * 64) {
    int i = tid >> 6, j = tid & 63;
    float s = 0.0f;
    for (int k = 0; k < 64; ++k) s += Wd[i * 64 + k] * Wf[(64 + k) * 64 + j];
    Wfused[tid] = s;
  }
  if (tid < 64) {
    float s = bf[tid];
    for (int k = 0; k < 64; ++k)
      s += bias_gat[k] * Wf[k * 64 + tid] + bd[k] * Wf[(64 + k) * 64 + tid];
    cvec[tid] = s;
  }
}

// ---------------------------------------------------------------------------
// helper: interleave two weight rows (2*kp, 2*kp+1) into pair layout
// sWp[kp][c][j] = W[2*kp+j][c]  -> B fragment {W[kk][n], W[kk+1][n]} is one b64
// ---------------------------------------------------------------------------
__device__ __forceinline__ void fill_pair_tile(const float* __restrict__ src,
                                               size_t row_stride,
                                               float (*dst)[64][2], int tid,
                                               int nthreads) {
  for (int i = tid; i < 512; i += nthreads) {  // 32 kp * 16 c4
    int kp = i >> 4, c4 = i & 15;
    const float* p0 = src + (size_t)(2 * kp) * row_stride + c4 * 4;
    float4 r0 = *(const float4*)p0;
    float4 r1 = *(const float4*)(p0 + row_stride);
    *(float4*)&dst[kp][c4 * 4 + 0][0] = make_float4(r0.x, r1.x, r0.y, r1.y);
    *(float4*)&dst[kp][c4 * 4 + 2][0] = make_float4(r0.z, r1.z, r0.w, r1.w);
  }
}

// ---------------------------------------------------------------------------
// out[M x 64] = in[M x 64] @ W[64 x 64] + bias  (WMMA f32 16x16x4)
// block = 256 threads (8 waves), 128 rows per block
// ---------------------------------------------------------------------------
__global__ __launch_bounds__(256) void gemm64_bias_kernel(
    const float* __restrict__ A, const float* __restrict__ W,
    const float* __restrict__ bias, float* __restrict__ out, int M) {
  __shared__ __align__(16) float sA[128][64];
  __shared__ __align__(16) float sWp[32][64][2];
  const int tid = threadIdx.x;
  const int row0 = blockIdx.x * 128;

  for (int i = tid; i < 2048; i += 256) {  // 128*64 floats as float4
    int r = i >> 4, c4 = i & 15;
    *(float4*)&sA[r][c4 * 4] = ((const float4*)(A + (size_t)(row0 + r) * 64))[c4];
  }
  fill_pair_tile(W, 64, sWp, tid, 256);
  __syncthreads();

  const int wave = tid >> 5, lane = tid & 31;
  const int lhalf = lane >> 4, lmod = lane & 15;
  const int wr0 = wave * 16;

  v8f acc[4];
#pragma unroll
  for (int t = 0; t < 4; ++t) acc[t] = (v8f){0, 0, 0, 0, 0, 0, 0, 0};

#pragma unroll
  for (int k0 = 0; k0 < 64; k0 += 4) {
    int kk = k0 + 2 * lhalf;
    v2f a = *(const v2f*)&sA[wr0 + lmod][kk];
    int kp = kk >> 1;
#pragma unroll
    for (int t = 0; t < 4; ++t) {
      v2f b = *(const v2f*)&sWp[kp][t * 16 + lmod][0];
      acc[t] = wmma_f32_k4(a, b, acc[t]);
    }
  }
#pragma unroll
  for (int t = 0; t < 4; ++t) {
    float bv = bias[t * 16 + lmod];
#pragma unroll
    for (int r = 0; r < 8; ++r) {
      int row = row0 + wr0 + r + 8 * lhalf;
      out[(size_t)row * 64 + t * 16 + lmod] = acc[t][r] + bv;
    }
  }
}

// ---------------------------------------------------------------------------
// dense_agg[b,n,c] += sum_m adj[n,m] * x[b,m,c]   (K-split, fp32 WMMA,
// double-buffered LDS: load chunk i+1 while computing chunk i)
// grid = (N/64, KSPLIT); block = 256 threads = 8 waves
// wave w: rows 16*(w>>1), batch w&1, all 64 cols
// ---------------------------------------------------------------------------
#define KSPLIT 4
__global__ __launch_bounds__(256) void adj_gemm_kernel(
    const float* __restrict__ adj, const float* __restrict__ x,
    float* __restrict__ dense_agg, int N) {
  __shared__ __align__(16) float sA[2][64][64];          // 32 KB
  __shared__ __align__(16) float sBp[2][2][32][64][2];   // 64 KB
  const int tid = threadIdx.x;
  const int row0 = blockIdx.x * 64;
  const int kbeg = blockIdx.y * (N / KSPLIT);
  const int nchunks = (N / KSPLIT) / 64;

  const int wave = tid >> 5, lane = tid & 31;
  const int lhalf = lane >> 4, lmod = lane & 15;
  const int wrow = (wave >> 1) * 16;  // 0,16,32,48
  const int wb = wave & 1;            // batch

  v8f acc[4];
#pragma unroll
  for (int t = 0; t < 4; ++t) acc[t] = (v8f){0, 0, 0, 0, 0, 0, 0, 0};

  auto load_chunk = [&](int buf, int kc) {
    // adj tile 64 rows x 64 k (row-major, A operand: pairs contiguous in k)
    for (int i = tid; i < 1024; i += 256) {
      int r = i >> 4, c4 = i & 15;
      const float* src = adj + (size_t)(row0 + r) * N + kc;
      *(float4*)&sA[buf][r][c4 * 4] = ((const float4*)src)[c4];
      __builtin_prefetch(src + 128, 0, 1);  // global_prefetch_b8, stream ahead
    }
    // x chunk, both batches, pair-interleaved for b64 B-fragment loads
    for (int i = tid; i < 1024; i += 256) {
      int b = i >> 9, j = i & 511;
      int kp = j >> 4, c4 = j & 15;
      const float* p0 = x + ((size_t)b * N + kc + 2 * kp) * 64 + c4 * 4;
      float4 r0 = *(const float4*)p0;
      float4 r1 = *(const float4*)(p0 + 64);
      *(float4*)&sBp[buf][b][kp][c4 * 4 + 0][0] =
          make_float4(r0.x, r1.x, r0.y, r1.y);
      *(float4*)&sBp[buf][b][kp][c4 * 4 + 2][0] =
          make_float4(r0.z, r1.z, r0.w, r1.w);
    }
  };

  load_chunk(0, kbeg);
  __syncthreads();

  for (int ci = 0; ci < nchunks; ++ci) {
    const int cur = ci & 1;
    if (ci + 1 < nchunks) load_chunk(cur ^ 1, kbeg + (ci + 1) * 64);
#pragma unroll
    for (int k0 = 0; k0 < 64; k0 += 4) {
      int kk = k0 + 2 * lhalf;
      v2f a = *(const v2f*)&sA[cur][wrow + lmod][kk];
      int kp = kk >> 1;
#pragma unroll
      for (int t = 0; t < 4; ++t) {
        v2f b = *(const v2f*)&sBp[cur][wb][kp][t * 16 + lmod][0];
        acc[t] = wmma_f32_k4(a, b, acc[t]);
      }
    }
    __syncthreads();
  }

#pragma unroll
  for (int t = 0; t < 4; ++t)
#pragma unroll
    for (int r = 0; r < 8; ++r) {
      int row = row0 + wrow + r + 8 * lhalf;
      atomicAdd(&dense_agg[((size_t)wb * N + row) * 64 + t * 16 + lmod],
                acc[t][r]);
    }
}

// ---------------------------------------------------------------------------
// edge phase 1: logits + segment max (atomic)
// ---------------------------------------------------------------------------
__global__ void edge_logits_kernel(const float* __restrict__ xl,
                                   const float* __restrict__ xr,
                                   const int* __restrict__ ei,
                                   const float* __restrict__ att,
                                   float* __restrict__ logits,
                                   float* __restrict__ mx, int N, int E,
                                   int Et) {
  int tid = blockIdx.x * blockDim.x + threadIdx.x;
  int total = 2 * Et;
  if (tid >= total) return;
  int b = tid / Et, e = tid - b * Et;
  int s, d;
  if (e < E) { s = ei[e]; d = ei[E + e]; } else { s = e - E; d = s; }
  const float* xs = xl + ((size_t)b * N + s) * 64;
  const float* xd = xr + ((size_t)b * N + d) * 64;
#pragma unroll
  for (int h = 0; h < HEADS; ++h) {
    float acc = 0.0f;
#pragma unroll
    for (int c = 0; c < HDIM; ++c) {
      float m = xs[h * HDIM + c] + xd[h * HDIM + c];
      float lr = m > 0.0f ? m : NEG_SLOPE * m;
      acc += att[h * HDIM + c] * lr;
    }
    logits[(size_t)tid * HEADS + h] = acc;
    atomicMaxF(&mx[((size_t)b * N + d) * HEADS + h], acc);
  }
}

// ---------------------------------------------------------------------------
// edge phase 2: ex = exp(logit - mx[dst]);  den[dst] += ex  (logits <- ex)
// ---------------------------------------------------------------------------
__global__ void edge_expden_kernel(const int* __restrict__ ei,
                                   float* __restrict__ logits,
                                   const float* __restrict__ mx,
                                   float* __restrict__ den, int N, int E,
                                   int Et) {
  int tid = blockIdx.x * blockDim.x + threadIdx.x;
  int total = 2 * Et;
  if (tid >= total) return;
  int b = tid / Et, e = tid - b * Et;
  int d = (e < E) ? ei[E + e] : (e - E);
#pragma unroll
  for (int h = 0; h < HEADS; ++h) {
    float lg = logits[(size_t)tid * HEADS + h];
    float m = mx[((size_t)b * N + d) * HEADS + h];
    float ex = expf(lg - m);
    logits[(size_t)tid * HEADS + h] = ex;
    atomicAdd(&den[((size_t)b * N + d) * HEADS + h], ex);
  }
}

// ---------------------------------------------------------------------------
// edge phase 3: gat[dst] += (ex/den[dst]) * xl[src]
// ---------------------------------------------------------------------------
__global__ void edge_agg_kernel(const float* __restrict__ xl,
                                const int* __restrict__ ei,
                                const float* __restrict__ logits,
                                const float* __restrict__ den,
                                float* __restrict__ gat, int N, int E, int Et) {
  int tid = blockIdx.x * blockDim.x + threadIdx.x;
  int total = 2 * Et;
  if (tid >= total) return;
  int b = tid / Et, e = tid - b * Et;
  int s, d;
  if (e < E) { s = ei[e]; d = ei[E + e]; } else { s = e - E; d = s; }
  const float* xs = xl + ((size_t)b * N + s) * 64;
  float* go = gat + ((size_t)b * N + d) * 64;
#pragma unroll
  for (int h = 0; h < HEADS; ++h) {
    float alpha = logits[(size_t)tid * HEADS + h] /
                  (den[((size_t)b * N + d) * HEADS + h] + 1e-16f);
#pragma unroll
    for (int c = 0; c < HDIM; ++c)
      atomicAdd(&go[h * HDIM + c], alpha * xs[h * HDIM + c]);
  }
}

// ---------------------------------------------------------------------------
// out = relu(gat_raw @ Wtop + dense_agg @ Wfused + cvec)   (WMMA f32)
// ---------------------------------------------------------------------------
__global__ __launch_bounds__(256) void final_kernel(
    const float* __restrict__ gat, const float* __restrict__ dense,
    const float* __restrict__ Wtop, const float* __restrict__ Wfused,
    const float* __restrict__ cvec, float* __restrict__ out, int M) {
  __shared__ __align__(16) float sA[128][64];
  __shared__ __align__(16) float sW1p[32][64][2];
  __shared__ __align__(16) float sW2p[32][64][2];
  const int tid = threadIdx.x;
  const int row0 = blockIdx.x * 128;

  fill_pair_tile(Wtop, 64, sW1p, tid, 256);
  fill_pair_tile(Wfused, 64, sW2p, tid, 256);

  const int wave = tid >> 5, lane = tid & 31;
  const int lhalf = lane >> 4, lmod = lane & 15;
  const int wr0 = wave * 16;

  v8f acc[4];
#pragma unroll
  for (int t = 0; t < 4; ++t) acc[t] = (v8f){0, 0, 0, 0, 0, 0, 0, 0};

  for (int pass = 0; pass < 2; ++pass) {
    const float* src = (pass == 0) ? gat : dense;
    __syncthreads();
    for (int i = tid; i < 2048; i += 256) {
      int r = i >> 4, c4 = i & 15;
      *(float4*)&sA[r][c4 * 4] =
          ((const float4*)(src + (size_t)(row0 + r) * 64))[c4];
    }
    __syncthreads();
#pragma unroll
    for (int k0 = 0; k0 < 64; k0 += 4) {
      int kk = k0 + 2 * lhalf;
      v2f a = *(const v2f*)&sA[wr0 + lmod][kk];
      int kp = kk >> 1;
#pragma unroll
      for (int t = 0; t < 4; ++t) {
        v2f b = (pass == 0) ? *(const v2f*)&sW1p[kp][t * 16 + lmod][0]
                            : *(const v2f*)&sW2p[kp][t * 16 + lmod][0];
        acc[t] = wmma_f32_k4(a, b, acc[t]);
      }
    }
  }
#pragma unroll
  for (int t = 0; t < 4; ++t) {
    float cv = cvec[t * 16 + lmod];
#pragma unroll
    for (int r = 0; r < 8; ++r) {
      int row = row0 + wr0 + r + 8 * lhalf;
      float v = acc[t][r] + cv;
      out[(size_t)row * 64 + t * 16 + lmod] = v > 0.0f ? v : 0.0f;
    }
  }
}

// ---------------------------------------------------------------------------
extern "C" void kernel_launch(void* const* d_in, const int* in_sizes, int n_in,
                              void* d_out, int out_size, void* d_ws,
                              size_t ws_size, hipStream_t stream) {
  const float* x        = (const float*)d_in[0];
  const int*   ei       = (const int*)d_in[1];
  const float* adj      = (const float*)d_in[2];
  const float* Wl       = (const float*)d_in[3];
  const float* bl       = (const float*)d_in[4];
  const float* Wr       = (const float*)d_in[5];
  const float* br       = (const float*)d_in[6];
  const float* att      = (const float*)d_in[7];
  const float* bias_gat = (const float*)d_in[8];
  const float* Wd       = (const float*)d_in[9];
  const float* bd       = (const float*)d_in[10];
  const float* Wf       = (const float*)d_in[11];
  const float* bf       = (const float*)d_in[12];

  const int N  = in_sizes[0] / 128;  // B*C = 2*64
  const int E  = in_sizes[1] / 2;
  const int Et = E + N;
  const int M  = 2 * N;  // B*N rows

  // workspace layout (floats): xl | xr | logits | mx | den | gat | dense | Wfused | cvec
  float* ws = (float*)d_ws;
  size_t o = 0;
  float* xl     = ws + o; o += (size_t)M * 64;
  float* xr     = ws + o; o += (size_t)M * 64;
  float* logits = ws + o; o += (size_t)2 * Et * HEADS;
  float* mx     = ws + o; o += (size_t)M * HEADS;
  float* den    = ws + o; o += (size_t)M * HEADS;   // den,gat,dense contiguous
  float* gat    = ws + o; o += (size_t)M * 64;
  float* dense  = ws + o; o += (size_t)M * 64;
  float* wfused = ws + o; o += 64 * 64;
  float* cvec   = ws + o; o += 64;

  const int nmx = M * HEADS;
  const int nz  = M * HEADS + M * 64 + M * 64;  // den + gat + dense
  {
    int mxn = nmx > nz ? nmx : nz;
    init_ws_kernel<<<(mxn + 255) / 256, 256, 0, stream>>>(mx, nmx, den, nz);
  }
  precompute_kernel<<<16, 256, 0, stream>>>(Wd, Wf, bias_gat, bd, bf, wfused,
                                            cvec);
  gemm64_bias_kernel<<<M / 128, 256, 0, stream>>>(x, Wl, bl, xl, M);
  gemm64_bias_kernel<<<M / 128, 256, 0, stream>>>(x, Wr, br, xr, M);

  const int etot = 2 * Et;
  edge_logits_kernel<<<(etot + 255) / 256, 256, 0, stream>>>(xl, xr, ei, att,
                                                             logits, mx, N, E, Et);
  edge_expden_kernel<<<(etot + 255) / 256, 256, 0, stream>>>(ei, logits, mx,
                                                             den, N, E, Et);
  edge_agg_kernel<<<(etot + 255) / 256, 256, 0, stream>>>(xl, ei, logits, den,
                                                          gat, N, E, Et);

  dim3 ag(N / 64, KSPLIT);
  adj_gemm_kernel<<<ag, 256, 0, stream>>>(adj, x, dense, N);

  final_kernel<<<M / 128, 256, 0, stream>>>(gat, dense, Wf /*top 64 rows*/,
                                            wfused, cvec, (float*)d_out, M);
}